// SpiralEnblock_68607807586562
// MI455X (gfx1250) — compile-verified
//
#include <hip/hip_runtime.h>

// ---------------------------------------------------------------------------
// SpiralEnblock for MI455X (gfx1250):
//   h[b,v,:] = elu( gather(x, spiral)[b,v,:] @ W^T + bias )   (WMMA f32 16x16x4)
//   out[b,row,:] += data * h[b,col,:]                          (atomic scatter)
//
// B=8, V=50000, K=9, CIN=64 -> KDIM=576, COUT=128, VOUT=12500, NNZ=37500
//
// MI455X strategy:
//   * full-FP32 matrix path (reference is f32): V_WMMA_F32_16X16X4_F32
//   * entire weight matrix (288KB) staged in LDS (320KB/WGP on CDNA5)
//   * one wave32 computes a 32(M) x 128(N) stripe:
//       - 16 independent v8f accumulators (two M-halves x 8 N-tiles)
//       - each LDS B-fragment feeds TWO WMMAs (A-lo, A-hi) -> 2:1 wmma:ds
//       - B fragments batched into bf[8] so ds_loads issue behind one wait
//   * h intermediate (204.8MB) lives in d_ws
// ---------------------------------------------------------------------------

typedef float v2f __attribute__((ext_vector_type(2)));
typedef float v8f __attribute__((ext_vector_type(8)));

#define B_    8
#define V_    50000
#define KSP_  9
#define CIN_  64
#define KDIM_ 576      // KSP_*CIN_
#define COUT_ 128
#define VOUT_ 12500
#define NNZ_  37500
#define MTILES_ 12500  // (B_*V_)/32

// ---------------------------------------------------------------------------
// Kernel 1: fused gather + GEMM + bias + ELU
// grid: 1563 blocks x 256 threads (8 waves); each wave: 32 rows x 128 cols
// dynamic LDS: 128*576*4 = 294912 bytes (full W)
// ---------------------------------------------------------------------------
__global__ void __launch_bounds__(256)
spiral_gemm_elu_kernel(const float* __restrict__ x,        // [B,V,CIN]
                       const float* __restrict__ weight,   // [COUT,KDIM]
                       const float* __restrict__ bias,     // [COUT]
                       const int*   __restrict__ spiral,   // [V,KSP]
                       float*       __restrict__ h)        // [B,V,COUT]
{
    extern __shared__ float ldsW[];            // [COUT_][KDIM_] row-major

    const int tid = threadIdx.x;

    // Cooperative stage of the full weight matrix into LDS (float4 = b128).
    {
        const float4* wsrc = (const float4*)weight;
        float4*       wdst = (float4*)ldsW;
        const int nvec = (COUT_ * KDIM_) / 4;  // 18432
        for (int i = tid; i < nvec; i += 256) wdst[i] = wsrc[i];
    }
    __syncthreads();

    const int wave  = tid >> 5;                      // 0..7
    const int lane  = tid & 31;
    const int mlane = lane & 15;                     // row-in-frag for A, col for B/C
    const int khalf = (lane >> 4) << 1;              // 0 (lanes 0-15) or 2 (lanes 16-31)

    const int mtile = blockIdx.x * 8 + wave;         // 32-row M tile
    if (mtile >= MTILES_) return;                    // wave-uniform tail guard
    const int row0  = mtile * 32;

    // Per-lane rows for the two A fragments. 50000 % 16 == 0, so each 16-row
    // fragment sits inside one batch; b/v computed per lane regardless.
    const int r0 = row0 + mlane;
    const int r1 = row0 + 16 + mlane;
    const int b0 = r0 / V_, v0 = r0 - b0 * V_;
    const int b1 = r1 / V_, v1 = r1 - b1 * V_;
    const float* xb0 = x + (size_t)b0 * V_ * CIN_;
    const float* xb1 = x + (size_t)b1 * V_ * CIN_;
    const int* srow0 = spiral + (size_t)v0 * KSP_;
    const int* srow1 = spiral + (size_t)v1 * KSP_;

    v8f accLo[8], accHi[8];
#pragma unroll
    for (int n = 0; n < 8; ++n) { accLo[n] = {}; accHi[n] = {}; }

    // K loop: 9 spiral neighbors x 64 input channels, 4 K per WMMA.
    for (int kb = 0; kb < KSP_; ++kb) {
        const int nbr0 = srow0[kb];
        const int nbr1 = srow1[kb];
        const float* xr0 = xb0 + (size_t)nbr0 * CIN_;
        const float* xr1 = xb1 + (size_t)nbr1 * CIN_;
        const int kbase = kb * CIN_;

#pragma unroll 2
        for (int kk = 0; kk < CIN_; kk += 4) {
            // Two A fragments (16x4 f32 each): lane holds A[mlane][khalf..+1]
            const v2f a0 = *(const v2f*)(xr0 + kk + khalf);   // global_load_b64
            const v2f a1 = *(const v2f*)(xr1 + kk + khalf);
            const int kcol = kbase + kk + khalf;

            // Batch all 8 B fragments first so ds_loads pipeline behind
            // a single s_wait_dscnt instead of one wait per WMMA.
            v2f bf[8];
#pragma unroll
            for (int n = 0; n < 8; ++n)
                bf[n] = *(const v2f*)(ldsW + (n * 16 + mlane) * KDIM_ + kcol);

#pragma unroll
            for (int n = 0; n < 8; ++n) {
                accLo[n] = __builtin_amdgcn_wmma_f32_16x16x4_f32(
                    false, a0, false, bf[n], (short)0, accLo[n], false, false);
                accHi[n] = __builtin_amdgcn_wmma_f32_16x16x4_f32(
                    false, a1, false, bf[n], (short)0, accHi[n], false, false);
            }
        }
    }

    // C layout (16x16 f32, 8 VGPRs): VGPR j -> M = j (lanes 0-15) or j+8
    // (lanes 16-31), N = mlane. Add bias, ELU, store h.
    const int moff = (lane >> 4) * 8;
#pragma unroll
    for (int n = 0; n < 8; ++n) {
        const int   col = n * 16 + mlane;
        const float bv  = bias[col];
#pragma unroll
        for (int j = 0; j < 8; ++j) {
            const int mLo = row0 + moff + j;
            const int mHi = row0 + 16 + moff + j;
            float vLo = accLo[n][j] + bv;
            float vHi = accHi[n][j] + bv;
            vLo = (vLo > 0.0f) ? vLo : expm1f(vLo);   // elu, alpha=1
            vHi = (vHi > 0.0f) ? vHi : expm1f(vHi);
            h[(size_t)mLo * COUT_ + col] = vLo;
            h[(size_t)mHi * COUT_ + col] = vHi;
        }
    }
}

// ---------------------------------------------------------------------------
// Kernel 2: COO scatter pool. One wave per (nnz, b) pair; 32 lanes x 4 chans.
// out must be pre-zeroed (hipMemsetAsync in kernel_launch).
// ---------------------------------------------------------------------------
__global__ void __launch_bounds__(256)
pool_scatter_kernel(const float* __restrict__ h,        // [B,V,COUT]
                    const int*   __restrict__ dt_row,   // [NNZ]
                    const int*   __restrict__ dt_col,   // [NNZ]
                    const float* __restrict__ dt_data,  // [NNZ]
                    float*       __restrict__ out)      // [B,VOUT,COUT]
{
    const int gwave = (blockIdx.x * blockDim.x + threadIdx.x) >> 5;
    const int lane  = threadIdx.x & 31;
    if (gwave >= NNZ_ * B_) return;

    const int nz  = gwave >> 3;      // / B_
    const int b   = gwave & 7;       // % B_
    const int row = dt_row[nz];
    const int col = dt_col[nz];
    const float d = dt_data[nz];

    const float4 hv =
        *(const float4*)(h + ((size_t)b * V_ + col) * COUT_ + lane * 4);
    float* op = out + ((size_t)b * VOUT_ + row) * COUT_ + lane * 4;

    atomicAdd(op + 0, d * hv.x);     // global_atomic_add_f32
    atomicAdd(op + 1, d * hv.y);
    atomicAdd(op + 2, d * hv.z);
    atomicAdd(op + 3, d * hv.w);
}

// ---------------------------------------------------------------------------
// Launch
// ---------------------------------------------------------------------------
extern "C" void kernel_launch(void* const* d_in, const int* in_sizes, int n_in,
                              void* d_out, int out_size, void* d_ws, size_t ws_size,
                              hipStream_t stream) {
    const float* x       = (const float*)d_in[0];   // [B,V,CIN]
    const float* weight  = (const float*)d_in[1];   // [COUT,KDIM]
    const float* bias    = (const float*)d_in[2];   // [COUT]
    const int*   spiral  = (const int*)  d_in[3];   // [V,KSP]
    const int*   dt_row  = (const int*)  d_in[4];   // [NNZ]
    const int*   dt_col  = (const int*)  d_in[5];   // [NNZ]
    const float* dt_data = (const float*)d_in[6];   // [NNZ]
    float*       out     = (float*)d_out;           // [B,VOUT,COUT]
    float*       h       = (float*)d_ws;            // [B,V,COUT] = 204.8 MB scratch

    (void)in_sizes; (void)n_in; (void)ws_size;

    // Stage 1: fused gather+GEMM+ELU. 12500 32-row tiles / 8 waves = 1562.5.
    const size_t lds_bytes = (size_t)COUT_ * KDIM_ * sizeof(float); // 294912
    spiral_gemm_elu_kernel<<<dim3((MTILES_ + 7) / 8), dim3(256), lds_bytes, stream>>>(
        x, weight, bias, spiral, h);

    // Stage 2: zero out, then atomic scatter pool.
    (void)hipMemsetAsync(d_out, 0, (size_t)out_size * sizeof(float), stream);
    const int nwaves = NNZ_ * B_;                    // 300000
    pool_scatter_kernel<<<dim3((nwaves + 7) / 8), dim3(256), 0, stream>>>(
        h, dt_row, dt_col, dt_data, out);
}